// PhraseEncodingMixin_56444460204501
// MI455X (gfx1250) — compile-verified
//
#include <hip/hip_runtime.h>

typedef __attribute__((ext_vector_type(2))) float v2f;
typedef __attribute__((ext_vector_type(4))) float v4f;
typedef __attribute__((ext_vector_type(8))) float v8f;

constexpr int B = 4, L = 256, H = 256;

// ---------------------------------------------------------------------------
// Phase 1: cumsum along L as a lower-triangular-ones matmul, in f32 WMMA.
//   c[b, j, h] = sum_{k<=j} x[b, k, h]  ==  (T @ X)[j, h],  T[j,k] = (k<=j)
// One wave per 16x16 (j,h) output tile; K-loop in steps of 4 using
// V_WMMA_F32_16X16X4_F32. A is synthesized in-register from the k<=j
// predicate (zero memory traffic for T); accumulation is full f32.
//
// VGPR layouts (ISA 7.12.2, 32-bit):
//   A 16x4 : lane = m + 16*(k>>1), vgpr = k&1         -> v2f per lane
//   B 4x16 : lane = n + 16*(k>>1), vgpr = k&1         -> v2f per lane
//   C 16x16: lane = n + 16*(m>=8), vgpr = m&7         -> v8f per lane
// ---------------------------------------------------------------------------
__global__ void cumsum_wmma(const float* __restrict__ x, float* __restrict__ c) {
  const int wave = threadIdx.x >> 5;
  const int lane = threadIdx.x & 31;
  const int tile = blockIdx.x * (blockDim.x >> 5) + wave;  // 0 .. B*16*16-1

  const int b  = tile >> 8;         // batch
  const int jt = (tile >> 4) & 15;  // j-tile (rows of C)
  const int ht = tile & 15;         // h-tile (cols of C)

  const int half = lane >> 4;       // 0: lanes 0-15, 1: lanes 16-31
  const int l16  = lane & 15;

  const int jrow = jt * 16 + l16;   // this lane's A-matrix row (global j)
  const int hcol = ht * 16 + l16;   // this lane's B/C column (global h)

  v8f acc = {0.f, 0.f, 0.f, 0.f, 0.f, 0.f, 0.f, 0.f};

  const int kend = jt * 16 + 16;    // only lower-triangular K-tiles contribute
  for (int k0 = 0; k0 < kend; k0 += 4) {
    const int ka = k0 + 2 * half;   // first of this lane's two k values

    v2f a;                          // T[jrow, ka], T[jrow, ka+1]
    a.x = (ka     <= jrow) ? 1.0f : 0.0f;
    a.y = (ka + 1 <= jrow) ? 1.0f : 0.0f;

    const float* xp = x + ((size_t)b * L + (size_t)ka) * H + hcol;
    v2f bv;                         // X[b, ka, hcol], X[b, ka+1, hcol]
    bv.x = xp[0];
    bv.y = xp[H];

    acc = __builtin_amdgcn_wmma_f32_16x16x4_f32(
        /*neg_a=*/false, a, /*neg_b=*/false, bv,
        /*c_mod=*/(short)0, acc, /*reuse_a=*/false, /*reuse_b=*/false);
  }

  // Write back: vgpr r holds row M = r + 8*half of the tile.
  float* cp = c + ((size_t)b * L + (size_t)(jt * 16 + 8 * half)) * H + hcol;
#pragma unroll
  for (int r = 0; r < 8; ++r) {
    cp[(size_t)r * H] = acc[r];
  }
}

// ---------------------------------------------------------------------------
// Phase 2: out[b,i,j,h] = (c[b,j,h] - c[b,i-1,h]) / (|i-j|+1)
// One 256-thread block per (b,i). cp is loaded once per thread; the 256
// reciprocals go through LDS once per block. Output is write-once, so use
// non-temporal 128-bit stores to keep the dead 256 MiB stream out of L2 and
// leave the 1 MiB cumsum array cache-resident for all 1024 blocks.
// ---------------------------------------------------------------------------
__global__ void span_mean(const float* __restrict__ c, float* __restrict__ out) {
  __shared__ float s_inv[L];

  const int bi = blockIdx.x;        // b*L + i
  const int b  = bi >> 8;
  const int i  = bi & (L - 1);
  const int t  = threadIdx.x;       // 256 threads

  int d = i - t; if (d < 0) d = -d;
  s_inv[t] = 1.0f / (float)(d + 1);
  __syncthreads();

  const int hv = t & 63;            // which float4 along H (64 * 4 = 256)
  const int jo = t >> 6;            // 0..3: j sub-offset per sweep

  const v4f* cb = (const v4f*)(c + (size_t)b * L * H);

  v4f cp;
  if (i == 0) cp = (v4f){0.f, 0.f, 0.f, 0.f};
  else        cp = cb[(size_t)(i - 1) * (H / 4) + hv];

  v4f* outp = (v4f*)(out + (size_t)bi * L * H);

  for (int jb = 0; jb < L; jb += 4) {
    const int j = jb + jo;
    const v4f  cj  = cb[(size_t)j * (H / 4) + hv];
    const float inv = s_inv[j];
    const v4f  r   = (cj - cp) * inv;
    __builtin_nontemporal_store(r, &outp[(size_t)j * (H / 4) + hv]);
  }
}

// ---------------------------------------------------------------------------
extern "C" void kernel_launch(void* const* d_in, const int* in_sizes, int n_in,
                              void* d_out, int out_size, void* d_ws, size_t ws_size,
                              hipStream_t stream) {
  (void)in_sizes; (void)n_in; (void)out_size; (void)ws_size;

  const float* x   = (const float*)d_in[0];  // [B, L, H] f32
  float*       out = (float*)d_out;          // [B, L, L, H] f32
  float*       c   = (float*)d_ws;           // cumsum scratch: B*L*H*4 = 1 MiB

  // Phase 1: B*16*16 = 1024 tiles, 8 waves (tiles) per 256-thread block.
  cumsum_wmma<<<(B * 16 * 16) / 8, 256, 0, stream>>>(x, c);

  // Phase 2: one block per (b, i); streams the 256 MiB output.
  span_mean<<<B * L, 256, 0, stream>>>(c, out);
}